// ForegroundSelectorForMask_20169166422183
// MI455X (gfx1250) — compile-verified
//
#include <hip/hip_runtime.h>
#include <hip/hip_bf16.h>

#define B_DIM 256
#define N_DIM 16384
#define K_FG  512
#define BLK   256
#define NW    (BLK / 32)

// gfx1250 async global->LDS path (ASYNCcnt-tracked). The builtin exists on this
// toolchain (round-1 diagnostic showed param type '__device__ int *'): it wants
// (addrspace(1) int*, addrspace(3) int*, imm offset, imm cpol).
#if defined(__gfx1250__) && __has_builtin(__builtin_amdgcn_global_load_async_to_lds_b32)
#define HAVE_ASYNC_LDS 1
#else
#define HAVE_ASYNC_LDS 0
#endif

#if HAVE_ASYNC_LDS
typedef __attribute__((address_space(1))) int g_int;
typedef __attribute__((address_space(3))) int l_int;
#endif

__device__ __forceinline__ void wait_async0() {
#if defined(__HIP_DEVICE_COMPILE__)
#if __has_builtin(__builtin_amdgcn_s_wait_asynccnt)
    __builtin_amdgcn_s_wait_asynccnt(0);
#else
    asm volatile("s_wait_asynccnt 0" ::: "memory");
#endif
#endif
}

__global__ __launch_bounds__(BLK) void fg_select_kernel(
    const int* __restrict__ class_targets,   // [B, N]
    const float* __restrict__ box_targets,   // [B, N, 4]
    const float* __restrict__ boxes,         // [B, N, 4]
    const int* __restrict__ plm,             // [B, N]
    float* __restrict__ out)                 // flat concat of 4 outputs
{
    __shared__ int s_fg[K_FG];
    __shared__ int s_bg[K_FG];
    __shared__ int s_cfg[NW];
    __shared__ int s_cbg[NW];
#if HAVE_ASYNC_LDS
    __shared__ int s_ct[2][BLK];
#endif

    const int b   = blockIdx.x;
    const int tid = threadIdx.x;
    const int wv  = tid >> 5;
    const int ln  = tid & 31;
    const unsigned lt = (1u << ln) - 1u;   // lanes below me (wave32)

    const int* row_ct = class_targets + (size_t)b * N_DIM;

    int fg_total = 0;
    int bg_total = 0;

#if HAVE_ASYNC_LDS
    // Prime chunk 0 into LDS asynchronously.
    __builtin_amdgcn_global_load_async_to_lds_b32(
        (g_int*)(void*)(row_ct + tid), (l_int*)&s_ct[0][tid], 0, 0);
    wait_async0();
    int buf = 0;
#endif

    // ---- Pass 1: stable stream compaction with early exit ----
    for (int base = 0; base < N_DIM; base += BLK) {
        const int i = base + tid;
#if HAVE_ASYNC_LDS
        const int ct = s_ct[buf][tid];
        const int nxt = base + BLK;
        if (nxt < N_DIM) {
            __builtin_amdgcn_global_load_async_to_lds_b32(
                (g_int*)(void*)(row_ct + nxt + tid), (l_int*)&s_ct[buf ^ 1][tid], 0, 0);
        }
#else
        const int ct = row_ct[i];
        if (base + BLK < N_DIM) __builtin_prefetch(row_ct + i + BLK, 0, 0);
#endif
        const bool fg = ct > 0;
#if __has_builtin(__builtin_amdgcn_ballot_w32)
        const unsigned m = __builtin_amdgcn_ballot_w32(fg);
#else
        const unsigned m = (unsigned)__ballot(fg);
#endif
        const int cfg = __popc(m);
        const int pfg = __popc(m & lt);
        const int pbg = ln - pfg;            // all 32 lanes active
        if (ln == 0) { s_cfg[wv] = cfg; s_cbg[wv] = 32 - cfg; }
        __syncthreads();

        int fgb = fg_total, bgb = bg_total, sf = 0, sb = 0;
#pragma unroll
        for (int w = 0; w < NW; ++w) {
            const int cf = s_cfg[w];
            const int cb = s_cbg[w];
            if (w < wv) { fgb += cf; bgb += cb; }
            sf += cf;
            sb += cb;
        }
        const int fpos = fgb + pfg;
        const int bpos = bgb + pbg;
        if (fg  && fpos < K_FG) s_fg[fpos] = i;
        if (!fg && bpos < K_FG) s_bg[bpos] = i;

        fg_total += sf;                      // identical in every thread
        bg_total += sb;
        if (fg_total >= K_FG) break;         // uniform early exit (~3 chunks expected)
        __syncthreads();                     // protect s_cfg/s_cbg reuse
#if HAVE_ASYNC_LDS
        wait_async0();                       // next chunk landed in LDS
        buf ^= 1;
#endif
    }
    __syncthreads();                         // s_fg/s_bg visible to all waves

    const int nfg = fg_total < K_FG ? fg_total : K_FG;

    // ---- Pass 2: gather + write (flat concat layout) ----
    const size_t O1 = (size_t)B_DIM * K_FG;            // fg_box_targets
    const size_t O2 = O1 + (size_t)B_DIM * K_FG * 4;   // fg_boxes
    const size_t O3 = O2 + (size_t)B_DIM * K_FG * 4;   // fg_label_map
    const float* bt_row = box_targets + (size_t)b * N_DIM * 4;
    const float* bx_row = boxes       + (size_t)b * N_DIM * 4;
    const int*   pl_row = plm         + (size_t)b * N_DIM;

    for (int k = tid; k < K_FG; k += BLK) {
        const int idx = (k < nfg) ? s_fg[k] : s_bg[k - nfg];
        const size_t rk = (size_t)b * K_FG + k;
        out[rk]      = (float)row_ct[idx];
        out[O3 + rk] = (float)pl_row[idx];
        const float4 t = *(const float4*)(bt_row + (size_t)idx * 4);  // global_load_b128
        const float4 x = *(const float4*)(bx_row + (size_t)idx * 4);
        *(float4*)(out + O1 + rk * 4) = t;
        *(float4*)(out + O2 + rk * 4) = x;
    }
}

extern "C" void kernel_launch(void* const* d_in, const int* in_sizes, int n_in,
                              void* d_out, int out_size, void* d_ws, size_t ws_size,
                              hipStream_t stream) {
    (void)in_sizes; (void)n_in; (void)out_size; (void)d_ws; (void)ws_size;
    const int*   ct = (const int*)  d_in[0];   // class_targets [B,N] int32
    const float* bt = (const float*)d_in[1];   // box_targets [B,N,4] f32
    const float* bx = (const float*)d_in[2];   // boxes [B,N,4] f32
    const int*   pm = (const int*)  d_in[3];   // proposal_to_label_map [B,N] int32
    float* out = (float*)d_out;
    hipLaunchKernelGGL(fg_select_kernel, dim3(B_DIM), dim3(BLK), 0, stream,
                       ct, bt, bx, pm, out);
}